// HeteroRGCNLayer_27350351741261
// MI455X (gfx1250) — compile-verified
//
#include <hip/hip_runtime.h>

typedef __attribute__((ext_vector_type(2))) float v2f;
typedef __attribute__((ext_vector_type(8))) float v8f;

#define D 128   // D_IN == D_OUT == 128

// ---------------------------------------------------------------------------
// Y[M,128] = X[M,128] @ W[128,128]^T + bias   via V_WMMA_F32_16X16X4_F32
// One wave computes one 16x16 output tile; 8 waves/block cover all 8 N-tiles
// of a single M-tile (shared X rows -> good L0/L2 reuse). M must be %16 == 0
// (NU = NI = 100000 = 6250*16).
//
// A-matrix (16x4 f32) per-lane layout: row = lane&15, k = kb*4 + (lane>>4)*2
// B-matrix (4x16)  = W^T tile: B[k][n] = W[n][k] -> same addressing over W rows.
// D-matrix: VGPR v, lane L -> (m = v + (L>>4)*8, n = L&15).
// ---------------------------------------------------------------------------
__global__ void gemm_wmma_f32(const float* __restrict__ X,
                              const float* __restrict__ W,
                              const float* __restrict__ bias,
                              float* __restrict__ Y) {
  const int lane = threadIdx.x & 31;
  const int wave = threadIdx.x >> 5;
  const int m0 = blockIdx.x * 16;
  const int n0 = wave * 16;
  const int r = lane & 15;
  const int koff = (lane >> 4) * 2;

  const float* arow = X + (size_t)(m0 + r) * D + koff;   // X rows of this M-tile
  const float* brow = W + (size_t)(n0 + r) * D + koff;   // W rows of this N-tile

  v8f c = {};
#pragma unroll 4
  for (int kb = 0; kb < D / 4; ++kb) {
    v2f a = *(const v2f*)(arow + kb * 4);   // global_load_b64
    v2f b = *(const v2f*)(brow + kb * 4);
    // 8 args: (neg_a, A, neg_b, B, c_mod, C, reuse_a, reuse_b)
    c = __builtin_amdgcn_wmma_f32_16x16x4_f32(false, a, false, b,
                                              (short)0, c, false, false);
  }

  const float bv = bias[n0 + r];           // column bias, same for all 8 rows
  const int hi = lane >> 4;
  float* yout = Y + (size_t)(m0 + hi * 8) * D + n0 + r;
#pragma unroll
  for (int v = 0; v < 8; ++v) yout[(size_t)v * D] = c[v] + bv;
}

// ---------------------------------------------------------------------------
// One wave per edge: gather Wh[src] (float4/lane = 128 floats) and atomically
// accumulate into sum[dst]; lane 0 bumps the per-dst edge count.
// Wh and sum are both L2-resident (51.2 MB each << 192 MB L2).
// ---------------------------------------------------------------------------
__global__ void scatter_edges(const float* __restrict__ Wh,
                              const int* __restrict__ src,
                              const int* __restrict__ dst,
                              float* __restrict__ sum,
                              float* __restrict__ cnt,
                              int nE) {
  const int lane = threadIdx.x & 31;
  const int e = blockIdx.x * (blockDim.x >> 5) + (threadIdx.x >> 5);
  if (e >= nE) return;
  const int s = src[e];
  const int d = dst[e];
  const float4 v = ((const float4*)(Wh + (size_t)s * D))[lane];  // global_load_b128
  float* dp = sum + (size_t)d * D + lane * 4;
  atomicAdd(dp + 0, v.x);
  atomicAdd(dp + 1, v.y);
  atomicAdd(dp + 2, v.z);
  atomicAdd(dp + 3, v.w);
  if (lane == 0) atomicAdd(cnt + d, 1.0f);
}

__global__ void zero_f32(float* __restrict__ p, size_t n) {
  size_t i = (size_t)blockIdx.x * blockDim.x + threadIdx.x;
  const size_t stride = (size_t)gridDim.x * blockDim.x;
  for (; i < n; i += stride) p[i] = 0.0f;
}

// h_user = sum_f / max(cnt_f,1) + sum_rb / max(cnt_rb,1)
__global__ void finalize_user(const float* __restrict__ sum_f,
                              const float* __restrict__ cnt_f,
                              const float* __restrict__ sum_rb,
                              const float* __restrict__ cnt_rb,
                              float* __restrict__ out, size_t n) {
  size_t i = (size_t)blockIdx.x * blockDim.x + threadIdx.x;
  if (i >= n) return;
  const size_t u = i >> 7;  // /128
  out[i] = sum_f[i] / fmaxf(cnt_f[u], 1.0f) + sum_rb[i] / fmaxf(cnt_rb[u], 1.0f);
}

// h_item: divide accumulated sums (already in d_out) by max(cnt,1), in place
__global__ void finalize_item(const float* __restrict__ cnt,
                              float* __restrict__ out, size_t n) {
  size_t i = (size_t)blockIdx.x * blockDim.x + threadIdx.x;
  if (i >= n) return;
  const size_t u = i >> 7;
  out[i] = out[i] / fmaxf(cnt[u], 1.0f);
}

extern "C" void kernel_launch(void* const* d_in, const int* in_sizes, int n_in,
                              void* d_out, int out_size, void* d_ws, size_t ws_size,
                              hipStream_t stream) {
  (void)n_in; (void)out_size; (void)ws_size;

  const float* feat_user  = (const float*)d_in[0];
  const float* feat_item  = (const float*)d_in[1];
  const float* W_follows  = (const float*)d_in[2];
  const float* b_follows  = (const float*)d_in[3];
  const float* W_rates    = (const float*)d_in[4];
  const float* b_rates    = (const float*)d_in[5];
  const float* W_ratedby  = (const float*)d_in[6];
  const float* b_ratedby  = (const float*)d_in[7];
  const int* src_follows  = (const int*)d_in[8];
  const int* dst_follows  = (const int*)d_in[9];
  const int* src_rates    = (const int*)d_in[10];
  const int* dst_rates    = (const int*)d_in[11];
  const int* src_ratedby  = (const int*)d_in[12];
  const int* dst_ratedby  = (const int*)d_in[13];

  const int NU  = in_sizes[0] / D;
  const int NI  = in_sizes[1] / D;
  const int Ef  = in_sizes[8];
  const int Er  = in_sizes[10];
  const int Erb = in_sizes[12];

  float* out_user = (float*)d_out;
  float* out_item = out_user + (size_t)NU * D;

  // workspace layout (floats)
  float* ws = (float*)d_ws;
  const size_t nmax = (size_t)(NU > NI ? NU : NI) * D;
  float* whbuf  = ws;                        // reused for all 3 etypes
  float* sum_f  = whbuf + nmax;              // NU*D
  float* sum_rb = sum_f + (size_t)NU * D;    // NU*D
  float* cnt_f  = sum_rb + (size_t)NU * D;   // NU
  float* cnt_rb = cnt_f + NU;                // NU
  float* cnt_r  = cnt_rb + NU;               // NI

  // zero accumulators (sums + counts are contiguous) and item output half
  const size_t zcount = (size_t)NU * D * 2 + (size_t)NU * 2 + (size_t)NI;
  zero_f32<<<2048, 256, 0, stream>>>(sum_f, zcount);
  zero_f32<<<2048, 256, 0, stream>>>(out_item, (size_t)NI * D);

  // etype 'follows': user -> user
  gemm_wmma_f32<<<NU / 16, 256, 0, stream>>>(feat_user, W_follows, b_follows, whbuf);
  scatter_edges<<<(Ef + 7) / 8, 256, 0, stream>>>(whbuf, src_follows, dst_follows,
                                                  sum_f, cnt_f, Ef);
  // etype 'ratedby': item -> user (reuse whbuf; stream order serializes)
  gemm_wmma_f32<<<NI / 16, 256, 0, stream>>>(feat_item, W_ratedby, b_ratedby, whbuf);
  scatter_edges<<<(Erb + 7) / 8, 256, 0, stream>>>(whbuf, src_ratedby, dst_ratedby,
                                                   sum_rb, cnt_rb, Erb);
  // etype 'rates': user -> item (accumulate straight into d_out item half)
  gemm_wmma_f32<<<NU / 16, 256, 0, stream>>>(feat_user, W_rates, b_rates, whbuf);
  scatter_edges<<<(Er + 7) / 8, 256, 0, stream>>>(whbuf, src_rates, dst_rates,
                                                  out_item, cnt_r, Er);

  const size_t nu_elems = (size_t)NU * D;
  const size_t ni_elems = (size_t)NI * D;
  finalize_user<<<(unsigned)((nu_elems + 255) / 256), 256, 0, stream>>>(
      sum_f, cnt_f, sum_rb, cnt_rb, out_user, nu_elems);
  finalize_item<<<(unsigned)((ni_elems + 255) / 256), 256, 0, stream>>>(
      cnt_r, out_item, ni_elems);
}